// CombinedLoss_38448547234846
// MI455X (gfx1250) — compile-verified
//
#include <hip/hip_runtime.h>
#include <math.h>

// ---------------------------------------------------------------------------
// CombinedLoss for MI455X / gfx1250 (wave32, WMMA).
// DSSIM separable 11-tap Gaussian conv implemented with V_WMMA_F32_16X16X4_F32
// (fp32 in/out, K=4) as matmuls against a Toeplitz band matrix.
// ---------------------------------------------------------------------------

typedef __attribute__((ext_vector_type(2))) float v2f;
typedef __attribute__((ext_vector_type(8))) float v8f;

__device__ __forceinline__ v8f wmma_k4(v2f a, v2f b, v8f c) {
  // (neg_a, A, neg_b, B, c_mod, C, reuse_a, reuse_b)
  return __builtin_amdgcn_wmma_f32_16x16x4_f32(false, a, false, b, (short)0, c,
                                               false, false);
}

__device__ __forceinline__ float block_reduce_sum(float v, float* sbuf) {
  #pragma unroll
  for (int off = 16; off > 0; off >>= 1) v += __shfl_xor(v, off, 32);
  const int tid = threadIdx.x;
  const int wv = tid >> 5, ln = tid & 31;
  if (ln == 0) sbuf[wv] = v;
  __syncthreads();
  float r = 0.0f;
  if (tid == 0) {
    const int nw = blockDim.x >> 5;
    for (int i = 0; i < nw; ++i) r += sbuf[i];
  }
  __syncthreads();
  return r;  // valid on thread 0
}

// ---------------------------------------------------------------------------
// DSSIM kernel: one wave per 16x16 output tile. 96 images x 256 tiles = 24576
// waves, 4 waves/block -> 6144 blocks.
// ---------------------------------------------------------------------------
__global__ __launch_bounds__(128) void dssim_kernel(
    const float* __restrict__ pred, const float* __restrict__ targ,
    float* __restrict__ partial) {
  __shared__ float Sp[4][32][33];   // zero-padded 26x26 input tile (pred)
  __shared__ float St[4][32][33];   // zero-padded 26x26 input tile (target)
  __shared__ float Hb[4][32][17];   // horizontal-conv intermediate, per wave
  __shared__ float red[4];

  const int tid  = threadIdx.x;
  const int wv   = tid >> 5;
  const int lane = tid & 31;
  const int m16  = lane & 15;  // M (or N) index within 16
  const int g    = lane >> 4;  // lane-group: selects K pair {2g, 2g+1}

  // Tile assignment
  const int wgl  = blockIdx.x * 4 + wv;      // 0..24575
  const int img  = wgl >> 8;                 // b*3+c
  const int tile = wgl & 255;
  const int ty = tile >> 4, tx = tile & 15;
  const int y0 = ty * 16, x0 = tx * 16;
  const float* __restrict__ P = pred + (size_t)img * 65536;
  const float* __restrict__ T = targ + (size_t)img * 65536;

  __builtin_prefetch(P + y0 * 256 + x0, 0, 3);
  __builtin_prefetch(T + y0 * 256 + x0, 0, 3);

  // Normalized 1D Gaussian: softmax over g_i+g_j is exactly separable into
  // p_i * p_j with p = softmax(g), so 11x11 kernel == outer(w1d, w1d).
  float winv;
  {
    float s = 0.f;
    #pragma unroll
    for (int t = 0; t < 11; ++t) {
      float d = (float)(t - 5);
      s += __expf(-d * d * (1.0f / 4.5f));
    }
    winv = 1.0f / s;
  }

  // Per-lane Toeplitz fragments: tp[c].{x,y} = w[(4c+2g{+1}) - m16] gated to
  // the band 0..10.  Identical values serve as stage-1 B (w[k-n], n=lane&15)
  // and stage-2 A (w[k-m], m=lane&15).
  v2f tp[7];
  #pragma unroll
  for (int c7 = 0; c7 < 7; ++c7) {
    #pragma unroll
    for (int h = 0; h < 2; ++h) {
      const int d = 4 * c7 + 2 * g + h - m16;
      float wv_ = 0.0f;
      if (d >= 0 && d <= 10) {
        const float dd = (float)(d - 5);
        wv_ = winv * __expf(-dd * dd * (1.0f / 4.5f));
      }
      if (h == 0) tp[c7].x = wv_; else tp[c7].y = wv_;
    }
  }

  // Stage input tile (26x26 valid, zero-padded to 32x32); lane = column.
  for (int r = 0; r < 32; ++r) {
    float pv = 0.f, tv = 0.f;
    const int gy = y0 + r, gx = x0 + lane;
    if (r < 26 && lane < 26 && gy < 256 && gx < 256) {
      pv = P[gy * 256 + gx];
      tv = T[gy * 256 + gx];
    }
    Sp[wv][r][lane] = pv;
    St[wv][r][lane] = tv;
  }
  // All LDS traffic is wave-private; DS ops are in-order within a wave.

  v8f vq[5];

  #pragma unroll
  for (int q = 0; q < 5; ++q) {
    // -------- stage 1: horizontal conv, H[r, x] over 32 rows ---------------
    #pragma unroll
    for (int rt = 0; rt < 2; ++rt) {
      v8f h = {};
      const int row = rt * 16 + m16;  // A-matrix M index = image row
      #pragma unroll
      for (int c7 = 0; c7 < 7; ++c7) {
        const int kb = 4 * c7 + 2 * g;
        const float p0 = Sp[wv][row][kb],     p1 = Sp[wv][row][kb + 1];
        const float t0 = St[wv][row][kb],     t1 = St[wv][row][kb + 1];
        v2f a;
        switch (q) {
          case 0:  a.x = p0;      a.y = p1;      break;  // mu1 source
          case 1:  a.x = t0;      a.y = t1;      break;  // mu2 source
          case 2:  a.x = p0 * p0; a.y = p1 * p1; break;  // E[p^2] source
          case 3:  a.x = t0 * t0; a.y = t1 * t1; break;  // E[t^2] source
          default: a.x = p0 * t0; a.y = p1 * t1; break;  // E[p*t] source
        }
        h = wmma_k4(a, tp[c7], h);
      }
      // D layout: lane 0-15 -> M=vgpr, lane 16-31 -> M=vgpr+8 ; N = lane&15
      #pragma unroll
      for (int v = 0; v < 8; ++v)
        Hb[wv][rt * 16 + v + 8 * g][m16] = h[v];
    }

    // -------- stage 2: vertical conv, Toeplitz as A (no transpose of H) ----
    v8f vacc = {};
    #pragma unroll
    for (int c7 = 0; c7 < 7; ++c7) {
      const int kb = 4 * c7 + 2 * g;
      v2f bf;
      bf.x = Hb[wv][kb][m16];
      bf.y = Hb[wv][kb + 1][m16];
      vacc = wmma_k4(tp[c7], bf, vacc);
    }
    vq[q] = vacc;
  }

  // -------- SSIM epilogue on the 16x16 tile (masked to 246x246) ------------
  const float c1 = 1e-4f, c2 = 9e-4f, EPS = 1e-8f;
  float lsum = 0.0f;
  #pragma unroll
  for (int v = 0; v < 8; ++v) {
    const int y = y0 + v + 8 * g;
    const int x = x0 + m16;
    if (y < 246 && x < 246) {
      const float mu1 = vq[0][v], mu2 = vq[1][v];
      const float m1s = mu1 * mu1, m2s = mu2 * mu2, m12 = mu1 * mu2;
      const float s1 = fmaxf(vq[2][v] - m1s, 0.0f);
      const float s2 = fmaxf(vq[3][v] - m2s, 0.0f);
      const float s12 = vq[4][v] - m12;
      const float lum = (2.0f * m12 + c1) / (m1s + m2s + c1 + EPS);
      const float cs  = (2.0f * s12 + c2) / (s1 + s2 + c2 + EPS);
      lsum += fminf(fmaxf(lum * cs, -1.0f), 1.0f);
    }
  }
  const float r = block_reduce_sum(lsum, red);
  if (tid == 0) partial[blockIdx.x] = r;
}

// ---------------------------------------------------------------------------
// Eyes/mouth weighted L1: grid (256, B), 256 threads, 1 pixel/thread.
// ---------------------------------------------------------------------------
__global__ __launch_bounds__(256) void em_kernel(
    const float* __restrict__ pred, const float* __restrict__ targ,
    const float* __restrict__ lm, float* __restrict__ partial) {
  __shared__ float slm[136];
  __shared__ float red[8];
  const int tid = threadIdx.x;
  const int b = blockIdx.y;
  if (tid < 136) slm[tid] = lm[b * 136 + tid];
  __syncthreads();

  const int pix = blockIdx.x * 256 + tid;
  const int py = pix >> 8, px = pix & 255;
  float eyeM = 0.0f, mouthM = 0.0f;
  #pragma unroll
  for (int i = 36; i < 68; ++i) {
    int cx = (int)slm[2 * i];     cx = min(max(cx, 0), 255);
    int cy = (int)slm[2 * i + 1]; cy = min(max(cy, 0), 255);
    const float dx = (float)(px - cx), dy = (float)(py - cy);
    const float d = sqrtf(dx * dx + dy * dy);
    const float rg = fminf(fmaxf(1.0f - d * (1.0f / 15.0f), 0.0f), 1.0f);
    if (i < 48) eyeM = fmaxf(eyeM, rg); else mouthM = fmaxf(mouthM, rg);
  }
  const float prio = fminf(eyeM + mouthM, 1.0f);
  const float wgt = 1.0f + prio * 299.0f;
  float s = 0.0f;
  #pragma unroll
  for (int c = 0; c < 3; ++c) {
    const size_t idx = ((size_t)(b * 3 + c)) * 65536 + pix;
    s += fabsf(pred[idx] - targ[idx]);
  }
  const float r = block_reduce_sum(wgt * s, red);
  if (tid == 0) partial[blockIdx.y * 256 + blockIdx.x] = r;
}

// ---------------------------------------------------------------------------
// Gaze: bilinear 32x32 eye patches. grid (2 eyes, B), 256 threads.
// ---------------------------------------------------------------------------
__device__ __forceinline__ float bilin(const float* __restrict__ img, float gx,
                                       float gy) {
  const float px = fminf(fmaxf((gx + 1.0f) * 0.5f * 255.0f, 0.0f), 255.0f);
  const float py = fminf(fmaxf((gy + 1.0f) * 0.5f * 255.0f, 0.0f), 255.0f);
  const int x0 = (int)floorf(px), y0 = (int)floorf(py);
  const int x1 = min(x0 + 1, 255), y1 = min(y0 + 1, 255);
  const float wx = px - (float)x0, wy = py - (float)y0;
  const float v00 = img[y0 * 256 + x0], v01 = img[y0 * 256 + x1];
  const float v10 = img[y1 * 256 + x0], v11 = img[y1 * 256 + x1];
  return v00 * (1 - wx) * (1 - wy) + v01 * wx * (1 - wy) +
         v10 * (1 - wx) * wy + v11 * wx * wy;
}

__global__ __launch_bounds__(256) void gaze_kernel(
    const float* __restrict__ pred, const float* __restrict__ targ,
    const float* __restrict__ lm, float* __restrict__ partial) {
  __shared__ float rect[4];
  __shared__ float red[8];
  const int tid = threadIdx.x;
  const int b = blockIdx.y;
  const int eye = blockIdx.x;  // 0 = left (36..41), 1 = right (42..47)

  if (tid == 0) {
    const int base = (eye == 0) ? 36 : 42;
    float xmn = 1e30f, xmx = -1e30f, ymn = 1e30f, ymx = -1e30f;
    for (int k = 0; k < 6; ++k) {
      const float x = lm[b * 136 + (base + k) * 2];
      const float y = lm[b * 136 + (base + k) * 2 + 1];
      xmn = fminf(xmn, x); xmx = fmaxf(xmx, x);
      ymn = fminf(ymn, y); ymx = fmaxf(ymx, y);
    }
    const float wd = xmx - xmn, ht = ymx - ymn;
    float x1 = fminf(fmaxf(xmn - wd * 0.3f, 0.0f), 255.0f);
    float x2 = fminf(fmaxf(xmx + wd * 0.3f, 0.0f), 255.0f);
    float y1 = fminf(fmaxf(ymn - ht * 0.3f, 0.0f), 255.0f);
    float y2 = fminf(fmaxf(ymx + ht * 0.3f, 0.0f), 255.0f);
    const bool small = ((x2 - x1) < 2.0f) || ((y2 - y1) < 2.0f);
    if (small) {
      const float cx = 0.5f * (x1 + x2), cy = 0.5f * (y1 + y2);
      x1 = fmaxf(cx - 1.0f, 0.0f); x2 = fminf(x1 + 2.0f, 255.0f);
      y1 = fmaxf(cy - 1.0f, 0.0f); y2 = fminf(y1 + 2.0f, 255.0f);
    }
    rect[0] = x1 * (2.0f / 255.0f) - 1.0f;
    rect[1] = x2 * (2.0f / 255.0f) - 1.0f;
    rect[2] = y1 * (2.0f / 255.0f) - 1.0f;
    rect[3] = y2 * (2.0f / 255.0f) - 1.0f;
  }
  __syncthreads();
  const float xs = rect[0], xe = rect[1], ys = rect[2], ye = rect[3];

  float acc = 0.0f;
  for (int s = tid; s < 3072; s += 256) {  // C(3) x 32 x 32
    const int ci = s >> 10;
    const int rem = s & 1023;
    const int iy = rem >> 5, ix = rem & 31;
    const float tX = (float)ix * (1.0f / 31.0f);
    const float tY = (float)iy * (1.0f / 31.0f);
    const float gx = xs + tX * (xe - xs);
    const float gy = ys + tY * (ye - ys);
    const float* Pc = pred + ((size_t)(b * 3 + ci)) * 65536;
    const float* Tc = targ + ((size_t)(b * 3 + ci)) * 65536;
    acc += fabsf(bilin(Pc, gx, gy) - bilin(Tc, gx, gy));
  }
  const float r = block_reduce_sum(acc, red);
  if (tid == 0) partial[b * 2 + eye] = r;
}

// ---------------------------------------------------------------------------
// Final deterministic fixed-order reduction + loss combination.
// ws layout: [0,6144) dssim ; [6144,14336) em ; [14336,14400) gaze
// ---------------------------------------------------------------------------
__global__ __launch_bounds__(256) void final_kernel(
    const float* __restrict__ ws, float* __restrict__ out) {
  __shared__ float red[8];
  const int tid = threadIdx.x;
  float a = 0.f, b = 0.f, c = 0.f;
  for (int i = tid; i < 6144; i += 256) a += ws[i];
  for (int i = tid; i < 8192; i += 256) b += ws[6144 + i];
  for (int i = tid; i < 64;   i += 256) c += ws[14336 + i];
  const float S1 = block_reduce_sum(a, red);
  const float S2 = block_reduce_sum(b, red);
  const float S3 = block_reduce_sum(c, red);
  if (tid == 0) {
    const float dssim = 0.5f * (1.0f - S1 * (1.0f / 5809536.0f));  // 96*246^2
    const float em = S2 * (1.0f / 6291456.0f);                     // 32*3*256^2
    const float gz = 0.5f * S3 * (1.0f / 98304.0f);                // 32*3*1024
    out[0] = 1.0f * dssim + 1.0f * em + 1.0f * gz;
  }
}

// ---------------------------------------------------------------------------
extern "C" void kernel_launch(void* const* d_in, const int* in_sizes, int n_in,
                              void* d_out, int out_size, void* d_ws,
                              size_t ws_size, hipStream_t stream) {
  (void)in_sizes; (void)n_in; (void)out_size; (void)ws_size;
  const float* pred = (const float*)d_in[0];
  const float* targ = (const float*)d_in[1];
  const float* lm   = (const float*)d_in[2];
  float* out = (float*)d_out;
  float* ws  = (float*)d_ws;

  dssim_kernel<<<6144, 128, 0, stream>>>(pred, targ, ws);
  em_kernel<<<dim3(256, 32), 256, 0, stream>>>(pred, targ, lm, ws + 6144);
  gaze_kernel<<<dim3(2, 32), 256, 0, stream>>>(pred, targ, lm, ws + 14336);
  final_kernel<<<1, 256, 0, stream>>>(ws, out);
}